// BERT4ETH_PR_DATA_46067819217007
// MI455X (gfx1250) — compile-verified
//
#include <hip/hip_runtime.h>

// ---------------------------------------------------------------------------
// BERT4ETH adjacency accumulation:
//   data[e]   = values[e] * dot(features[e,0:5], w[0:5])
//   out[row[e]] += data[e]           (segment_sum over 3M nodes)
//
// Memory-bound: ~470 MB streamed (NT hints) at 23.3 TB/s => ~21us floor.
// 12 MB accumulator stays hot in the 192 MB L2; atomics are L2-local
// (no-return global_atomic_add_f32).
// dot4 part computed with V_WMMA_F32_16X16X4_F32 (A = weight broadcast to all
// rows, B = 16 edges' features) with layouts chosen so NO cross-lane shuffles
// are needed and each feature float is loaded exactly once.
// All offsets are 32-bit (E*5 fits int by construction) so the inner loop
// uses SADDR + 32-bit voffset addressing instead of 64-bit VALU adds.
// ---------------------------------------------------------------------------

typedef float v2f __attribute__((ext_vector_type(2)));
typedef float v8f __attribute__((ext_vector_type(8)));

__device__ __forceinline__ float nt_load(const float* p) {
    return __builtin_nontemporal_load(p);
}
__device__ __forceinline__ int nt_load(const int* p) {
    return __builtin_nontemporal_load(p);
}

__global__ void __launch_bounds__(256)
edge_segsum_wmma(const float* __restrict__ F,   // [E,5] features
                 const float* __restrict__ V,   // [E]   values
                 const float* __restrict__ W,   // [5]   a0_weight
                 const int*   __restrict__ R,   // [E]   rows
                 float*       __restrict__ out, // [num_nodes]
                 int E)
{
    // Broadcast weights (uniform address -> scalar loads).
    const float w0 = W[0], w1 = W[1], w2 = W[2], w3 = W[3], w4 = W[4];

    const unsigned lane = threadIdx.x & 31u;
    const bool     hi   = lane >= 16u;   // upper half-wave
    const unsigned n    = lane & 15u;    // column within 16-edge tile
    const unsigned kb   = hi ? 2u : 0u;  // K base for this half-wave

    // A-matrix (16x4 f32): all 16 rows = w[0..3].
    // ISA layout: VGPR0 = K=0 (lanes 0-15) / K=2 (lanes 16-31); VGPR1 = K=1 / K=3.
    v2f a;
    a.x = hi ? w2 : w0;
    a.y = hi ? w3 : w1;

    const unsigned wavesPerBlock = blockDim.x >> 5;
    const unsigned waveId   = blockIdx.x * wavesPerBlock + (threadIdx.x >> 5);
    const unsigned numWaves = gridDim.x * wavesPerBlock;
    const unsigned tail     = (unsigned)E & ~31u;   // 32-edge granularity
    const unsigned stride   = numWaves * 32u;

    // Wave-uniform loop bound: EXEC stays all-1s (WMMA requirement).
    for (unsigned base = waveId * 32u; base < tail; base += stride) {
        const unsigned eMine = base + lane;            // edge this lane finalizes
        const unsigned f0    = (base + n) * 5u + kb;   // tile0 B element offset
        // tile1 is the same address + 320 bytes (16 edges * 20 B).

        // B-matrix fill (4x16 f32, mirrored K layout): lane reads its two
        // K-elements straight from memory; all 4 K's of all 32 edges are
        // covered exactly once across the wave.
        v2f b0, b1;
        b0.x = nt_load(F + f0);
        b0.y = nt_load(F + f0 + 1u);
        b1.x = nt_load(F + f0 + 80u);
        b1.y = nt_load(F + f0 + 81u);

        const float f4  = nt_load(F + eMine * 5u + 4u);
        const float val = nt_load(V + eMine);
        const int   row = nt_load(R + eMine);

        // Unconditional prefetch of the next chunk of the feature stream
        // (global_prefetch_b8, WGP scope). Speculative prefetch past the end
        // of the buffer is silently dropped by hardware.
        __builtin_prefetch(F + (base + stride + lane) * 5u, 0, 3);

        // D = A x B (+0): every row of D equals the per-column dot4, so each
        // lane's d[0] holds dot4 for edge (lane % 16) of its tile.
        v8f c = {};
        v8f d0 = __builtin_amdgcn_wmma_f32_16x16x4_f32(
            false, a, false, b0, (short)0, c, false, false);
        v8f d1 = __builtin_amdgcn_wmma_f32_16x16x4_f32(
            false, a, false, b1, (short)0, c, false, false);

        const float dot4 = hi ? d1[0] : d0[0];

        // L2-local f32 atomic scatter (no-return form).
        atomicAdd(out + row, val * (dot4 + w4 * f4));
    }

    // Scalar tail for E % 32 edges (dead for E = 2^24, kept for generality).
    for (unsigned e = tail + blockIdx.x * blockDim.x + threadIdx.x;
         e < (unsigned)E; e += gridDim.x * blockDim.x) {
        const float* f = F + e * 5u;
        const float dot = f[0]*w0 + f[1]*w1 + f[2]*w2 + f[3]*w3 + f[4]*w4;
        atomicAdd(out + R[e], V[e] * dot);
    }
}

extern "C" void kernel_launch(void* const* d_in, const int* in_sizes, int n_in,
                              void* d_out, int out_size, void* d_ws, size_t ws_size,
                              hipStream_t stream) {
    // setup_inputs order: features, values, a0_weight, rows, num_nodes
    const float* F = (const float*)d_in[0];
    const float* V = (const float*)d_in[1];
    const float* W = (const float*)d_in[2];
    const int*   R = (const int*)d_in[3];
    const int    E = in_sizes[1];   // values has E elements

    // Deterministic output: zero the 3M-float accumulator every call
    // (hipMemsetAsync is stream-ordered and graph-capturable).
    hipMemsetAsync(d_out, 0, (size_t)out_size * sizeof(float), stream);

    // 4096 blocks x 256 threads = 32768 waves; each wave grid-strides
    // 32-edge chunks (~16 iterations for E = 2^24).
    dim3 grid(4096), block(256);
    edge_segsum_wmma<<<grid, block, 0, stream>>>(F, V, W, R, (float*)d_out, E);
}